// GraphNorm_57604101374346
// MI455X (gfx1250) — compile-verified
//
#include <hip/hip_runtime.h>
#include <hip/hip_bf16.h>
#include <math.h>

typedef __attribute__((ext_vector_type(2))) float v2f;
typedef __attribute__((ext_vector_type(4))) float v4f;
typedef __attribute__((ext_vector_type(8))) float v8f;

#define UNITS 256
#define BATCH 64
#define ROWS_PER_BLOCK 256
#define GN_EPS 1e-5f

// ---------------------------------------------------------------------------
// Kernel 0: zero the sum/sumsq accumulators (d_ws is poisoned by the harness,
// and atomics accumulate, so this must run every call).
// ---------------------------------------------------------------------------
__global__ void gn_zero_kernel(float* __restrict__ ws, int count) {
    int i = blockIdx.x * blockDim.x + threadIdx.x;
    if (i < count) ws[i] = 0.0f;
}

// ---------------------------------------------------------------------------
// Kernel 1: per-(graph, unit) sum and sum-of-squares via WMMA.
//
// V_WMMA_F32_16X16X4_F32 computes D = A(16x4) * B(4x16) + C. With A = ones,
// D[m,n] = sum_k B[k,n] for every m: each WMMA folds a 4-row x 16-col tile of
// x into the per-column f32 accumulator. B layout (4x16 f32, 2 VGPRs/lane):
// lanes 0-15 hold rows K=0,1 of columns 0-15; lanes 16-31 hold rows K=2,3.
// So lane l loads x[r + 2*(l>=16)][colbase + (l&15)] and the row below it —
// two loads whose 32-lane footprint is four full 64B segments (fully
// coalesced). A second WMMA chain accumulates VALU-squared values for sumsq.
// One 256-thread block = 8 waves covers all 256 columns x 256 rows; partials
// are combined across the 8 blocks per graph with global f32 atomics.
// EXEC is all-ones throughout the WMMA loop (no divergence).
//
// x loads here use default (regular-temporal) policy on purpose: x (128 MB)
// fits in the 192 MB L2, so this pass warms L2 for the apply pass.
// ---------------------------------------------------------------------------
__global__ void __launch_bounds__(256)
gn_stats_wmma_kernel(const float* __restrict__ x,
                     float* __restrict__ sum,
                     float* __restrict__ sumsq,
                     int n, int blocksPerGraph) {
    const int graph = blockIdx.x / blocksPerGraph;
    const int rb    = blockIdx.x % blocksPerGraph;
    const int wave  = threadIdx.x >> 5;
    const int lane  = threadIdx.x & 31;

    const int ca   = wave * 32;          // first 16-column group of this wave
    const int cb   = ca + 16;            // second 16-column group
    const int nsub = lane & 15;          // column within group
    const int rsel = (lane >> 4) << 1;   // lanes 0-15 -> rows +0/+1, 16-31 -> +2/+3

    const v2f ones = {1.0f, 1.0f};
    v8f acc_sa = {}; // column sums, group a
    v8f acc_qa = {}; // column sumsq, group a
    v8f acc_sb = {}; // column sums, group b
    v8f acc_qb = {}; // column sumsq, group b

    const size_t gbase = (size_t)graph * (size_t)n * UNITS;
    const int r0 = rb * ROWS_PER_BLOCK;

    for (int r = 0; r < ROWS_PER_BLOCK; r += 4) {
        const int rlo = r0 + r + rsel;
        const float* pa = x + gbase + (size_t)rlo * UNITS + (ca + nsub);
        const float* pb = x + gbase + (size_t)rlo * UNITS + (cb + nsub);
        float a0 = pa[0];
        float a1 = pa[UNITS];
        float b0 = pb[0];
        float b1 = pb[UNITS];

        v2f Ba = {a0, a1};
        v2f Qa = {a0 * a0, a1 * a1};
        v2f Bb = {b0, b1};
        v2f Qb = {b0 * b0, b1 * b1};

        acc_sa = __builtin_amdgcn_wmma_f32_16x16x4_f32(false, ones, false, Ba,
                                                       (short)0, acc_sa, false, false);
        acc_qa = __builtin_amdgcn_wmma_f32_16x16x4_f32(false, ones, false, Qa,
                                                       (short)0, acc_qa, false, false);
        acc_sb = __builtin_amdgcn_wmma_f32_16x16x4_f32(false, ones, false, Bb,
                                                       (short)0, acc_sb, false, false);
        acc_qb = __builtin_amdgcn_wmma_f32_16x16x4_f32(false, ones, false, Qb,
                                                       (short)0, acc_qb, false, false);
    }

    // Every row m of D holds the same column sum, and lanes 0-15 / 16-31
    // duplicate columns; element 0 of the accumulator in lanes 0-15 is enough.
    if (lane < 16) {
        const int base = graph * UNITS;
        atomicAdd(&sum[base + ca + nsub],   acc_sa[0]);
        atomicAdd(&sumsq[base + ca + nsub], acc_qa[0]);
        atomicAdd(&sum[base + cb + nsub],   acc_sb[0]);
        atomicAdd(&sumsq[base + cb + nsub], acc_qb[0]);
    }
}

// ---------------------------------------------------------------------------
// Kernel 2: turn (sum, sumsq) into (mean, 1/(std+eps)). 16K threads total.
// std is population std (ddof=0), matching jnp.std / tf.reduce_std.
// ---------------------------------------------------------------------------
__global__ void gn_finalize_kernel(const float* __restrict__ sum,
                                   const float* __restrict__ sumsq,
                                   float* __restrict__ mean,
                                   float* __restrict__ invstd,
                                   int n, int count) {
    int i = blockIdx.x * blockDim.x + threadIdx.x;
    if (i >= count) return;
    const float inv_n = 1.0f / (float)n;
    float m = sum[i] * inv_n;
    float v = fmaxf(sumsq[i] * inv_n - m * m, 0.0f);
    mean[i]   = m;
    invstd[i] = 1.0f / (sqrtf(v) + GN_EPS);
}

// ---------------------------------------------------------------------------
// Kernel 3: apply. Streaming pass with cache-policy control:
//  - x: non-temporal load (last use; still hits L2 if resident from stats),
//  - gamma/beta: non-temporal loads (read exactly once — must not evict x),
//  - out: non-temporal store (never re-read),
//  - mean/invstd (128 KB total): regular temporal, heavy L1/L2 reuse.
// All bulk traffic is b128 per lane via clang ext_vector float4 (v4f), which
// the nontemporal builtins accept (HIP's struct float4 is not).
// ---------------------------------------------------------------------------
__global__ void __launch_bounds__(256)
gn_apply_kernel(const float* __restrict__ x,
                const float* __restrict__ gamma,
                const float* __restrict__ beta,
                const float* __restrict__ mean,
                const float* __restrict__ invstd,
                float* __restrict__ out,
                int n, size_t total4) {
    size_t i = (size_t)blockIdx.x * blockDim.x + threadIdx.x;
    if (i >= total4) return;
    const size_t base = i * 4;
    const int row   = (int)(base >> 8);       // UNITS == 256
    const int graph = row / n;
    const int col   = (int)(base & (UNITS - 1));
    const int sidx  = graph * UNITS + col;

    v4f xv = __builtin_nontemporal_load((const v4f*)(x + base));
    v4f g  = __builtin_nontemporal_load((const v4f*)(gamma + base));
    v4f b  = __builtin_nontemporal_load((const v4f*)(beta + base));
    v4f m  = *(const v4f*)(mean + sidx);
    v4f is = *(const v4f*)(invstd + sidx);

    v4f o;
    o[0] = fmaf(g[0], (xv[0] - m[0]) * is[0], b[0]);
    o[1] = fmaf(g[1], (xv[1] - m[1]) * is[1], b[1]);
    o[2] = fmaf(g[2], (xv[2] - m[2]) * is[2], b[2]);
    o[3] = fmaf(g[3], (xv[3] - m[3]) * is[3], b[3]);
    __builtin_nontemporal_store(o, (v4f*)(out + base));
}

// ---------------------------------------------------------------------------
// Host-side launch. Inputs: x (N*U f32), gamma (N*U f32), beta (N*U f32),
// batch_size (scalar, fixed at 64 per the reference setup). U is 256 by the
// reference; N is derived from in_sizes[0].
// ---------------------------------------------------------------------------
extern "C" void kernel_launch(void* const* d_in, const int* in_sizes, int n_in,
                              void* d_out, int out_size, void* d_ws, size_t ws_size,
                              hipStream_t stream) {
    const float* x     = (const float*)d_in[0];
    const float* gamma = (const float*)d_in[1];
    const float* beta  = (const float*)d_in[2];
    float* out = (float*)d_out;

    const int NU = in_sizes[0];
    const int N  = NU / UNITS;
    const int n  = N / BATCH;              // rows per graph (2048)
    const int statCount = BATCH * UNITS;   // 16384

    float* ws     = (float*)d_ws;
    float* sum    = ws;
    float* sumsq  = ws + statCount;
    float* mean   = ws + 2 * statCount;
    float* invstd = ws + 3 * statCount;

    // 0) zero accumulators (sum and sumsq are contiguous)
    gn_zero_kernel<<<(2 * statCount + 255) / 256, 256, 0, stream>>>(sum, 2 * statCount);

    // 1) WMMA stats (also warms L2 with x)
    const int blocksPerGraph = (n + ROWS_PER_BLOCK - 1) / ROWS_PER_BLOCK;  // 8
    gn_stats_wmma_kernel<<<BATCH * blocksPerGraph, 256, 0, stream>>>(
        x, sum, sumsq, n, blocksPerGraph);

    // 2) finalize mean / invstd
    gn_finalize_kernel<<<(statCount + 255) / 256, 256, 0, stream>>>(
        sum, sumsq, mean, invstd, n, statCount);

    // 3) normalize + affine, vectorized 4 floats/thread, NT cache policy
    const size_t total4 = (size_t)NU / 4;
    gn_apply_kernel<<<(unsigned)((total4 + 255) / 256), 256, 0, stream>>>(
        x, gamma, beta, mean, invstd, out, n, total4);
}